// DriverNet_35751307772400
// MI455X (gfx1250) — compile-verified
//
#include <hip/hip_runtime.h>

typedef __attribute__((ext_vector_type(2))) float v2f;
typedef __attribute__((ext_vector_type(8))) float v8f;

#define IN_DIM 21
#define HID 5
#define NGATE 20 /* 4*HID */

// LDS layout offsets (floats)
#define OFF_WHH0 0
#define OFF_WIH1 100
#define OFF_WHH1 200
#define OFF_WLIN 300
#define OFF_B0   305
#define OFF_B1   325
#define OFF_BLIN 345
#define SMEM_N   346

__device__ __forceinline__ float fast_sigmoid(float x) {
    return 1.0f / (1.0f + __expf(-x));
}
// tanh(x) = 1 - 2/(exp(2x)+1): exact limits at +/-inf, single v_exp_f32
__device__ __forceinline__ float fast_tanh(float x) {
    return 1.0f - 2.0f / (__expf(2.0f * x) + 1.0f);
}

__global__ __launch_bounds__(256) void lstm2_wmma_kernel(
    const float* __restrict__ x,    const float* __restrict__ h0,
    const float* __restrict__ c0,
    const float* __restrict__ Wih0, const float* __restrict__ Whh0,
    const float* __restrict__ bih0, const float* __restrict__ bhh0,
    const float* __restrict__ Wih1, const float* __restrict__ Whh1,
    const float* __restrict__ bih1, const float* __restrict__ bhh1,
    const float* __restrict__ Wlin, const float* __restrict__ blin,
    float* __restrict__ out, int Bn)
{
    __shared__ float smem[SMEM_N];
    const int tid = threadIdx.x;

    // ---- one-time cooperative LDS fill of tiny weights/biases ----
    for (int i = tid; i < 100; i += 256) {
        smem[OFF_WHH0 + i] = Whh0[i];
        smem[OFF_WIH1 + i] = Wih1[i];
        smem[OFF_WHH1 + i] = Whh1[i];
    }
    if (tid < HID) smem[OFF_WLIN + tid] = Wlin[tid];
    if (tid >= 32 && tid < 32 + NGATE) smem[OFF_B0 + tid - 32] = bih0[tid - 32] + bhh0[tid - 32];
    if (tid >= 64 && tid < 64 + NGATE) smem[OFF_B1 + tid - 64] = bih1[tid - 64] + bhh1[tid - 64];
    if (tid == 96) smem[OFF_BLIN] = blin[0];
    __syncthreads();

    const int lane = tid & 31;
    const int wave = tid >> 5;
    const int n    = lane & 15;  // M (gate) / N (sample) index within fragment
    const int hi   = lane >> 4;  // half select: K offset 0 or 2 (ISA 16x4/4x16 layout)

    // ---- A fragments: W_ih0 rows (gates) x 4-wide K chunks, K padded 21->24 ----
    v2f a0[6], a1[6];
#pragma unroll
    for (int kc = 0; kc < 6; ++kc) {
        const int col = kc * 4 + hi * 2;
        a0[kc].x = (col     < IN_DIM) ? Wih0[n * IN_DIM + col]     : 0.0f;
        a0[kc].y = (col + 1 < IN_DIM) ? Wih0[n * IN_DIM + col + 1] : 0.0f;
        const int m2 = 16 + n; // gates 16..19, rows 20..31 zero-padded
        a1[kc].x = (m2 < NGATE && col     < IN_DIM) ? Wih0[m2 * IN_DIM + col]     : 0.0f;
        a1[kc].y = (m2 < NGATE && col + 1 < IN_DIM) ? Wih0[m2 * IN_DIM + col + 1] : 0.0f;
    }

    const long waveBase = ((long)blockIdx.x * 8 + wave) * 32; // 32 samples per wave
    float g20[NGATE];

#pragma unroll
    for (int sg = 0; sg < 2; ++sg) {
        const long sBase = waveBase + sg * 16;
        v8f acc0 = {}; // gates 0..15 x samples 0..15
        v8f acc1 = {}; // gates 16..19 (padded to 32)
#pragma unroll
        for (int kc = 0; kc < 6; ++kc) {
            const int col = kc * 4 + hi * 2;
            const float* xr = x + (size_t)(sBase + n) * IN_DIM;
            v2f bf;
            bf.x = (col     < IN_DIM) ? xr[col]     : 0.0f;
            bf.y = (col + 1 < IN_DIM) ? xr[col + 1] : 0.0f;
            // D = A(16x4) * B(4x16) + C  -- two independent accumulator chains
            acc0 = __builtin_amdgcn_wmma_f32_16x16x4_f32(false, a0[kc], false, bf,
                                                         (short)0, acc0, false, false);
            acc1 = __builtin_amdgcn_wmma_f32_16x16x4_f32(false, a1[kc], false, bf,
                                                         (short)0, acc1, false, false);
        }
        // D layout: elem r, lanes 0-15 -> gate r, sample=lane; lanes 16-31 -> gate r+8.
        // Redistribute so lane (sg*16 + n) owns all 20 gates of its sample.
        const bool mine = (sg == 0) ? (lane < 16) : (lane >= 16);
#pragma unroll
        for (int r = 0; r < 8; ++r) {
            float lo = __shfl(acc0[r], n, 32);
            float hv = __shfl(acc0[r], n + 16, 32);
            if (mine) { g20[r] = lo; g20[8 + r] = hv; }
        }
#pragma unroll
        for (int r = 0; r < 4; ++r) {
            float t = __shfl(acc1[r], n, 32);
            if (mine) g20[16 + r] = t;
        }
    }

    // ---- scalar epilogue: one lane = one sample ----
    const long s = waveBase + lane;
    float h0a[HID], c0a[HID], h0b[HID], c0b[HID];
    const float* ph0a = h0 + (size_t)s * HID;
    const float* ph0b = h0 + (size_t)Bn * HID + (size_t)s * HID;
    const float* pc0a = c0 + (size_t)s * HID;
    const float* pc0b = c0 + (size_t)Bn * HID + (size_t)s * HID;
#pragma unroll
    for (int k = 0; k < HID; ++k) {
        h0a[k] = ph0a[k]; h0b[k] = ph0b[k];
        c0a[k] = pc0a[k]; c0b[k] = pc0b[k];
    }

    // layer 0: add hidden path + combined bias, then gate nonlinearity
#pragma unroll
    for (int gi = 0; gi < NGATE; ++gi) {
        float acc = g20[gi] + smem[OFF_B0 + gi];
#pragma unroll
        for (int k = 0; k < HID; ++k)
            acc = fmaf(smem[OFF_WHH0 + gi * HID + k], h0a[k], acc);
        g20[gi] = acc;
    }
    float h1v[HID];
#pragma unroll
    for (int k = 0; k < HID; ++k) {
        float iv = fast_sigmoid(g20[k]);
        float fv = fast_sigmoid(g20[HID + k]);
        float gv = fast_tanh(g20[2 * HID + k]);
        float ov = fast_sigmoid(g20[3 * HID + k]);
        float cn = fv * c0a[k] + iv * gv;
        h1v[k] = ov * fast_tanh(cn);
    }

    // layer 1 (K=5, pure VALU)
    float t1[NGATE];
#pragma unroll
    for (int gi = 0; gi < NGATE; ++gi) {
        float acc = smem[OFF_B1 + gi];
#pragma unroll
        for (int k = 0; k < HID; ++k) {
            acc = fmaf(smem[OFF_WIH1 + gi * HID + k], h1v[k], acc);
            acc = fmaf(smem[OFF_WHH1 + gi * HID + k], h0b[k], acc);
        }
        t1[gi] = acc;
    }
    float h2v[HID];
#pragma unroll
    for (int k = 0; k < HID; ++k) {
        float iv = fast_sigmoid(t1[k]);
        float fv = fast_sigmoid(t1[HID + k]);
        float gv = fast_tanh(t1[2 * HID + k]);
        float ov = fast_sigmoid(t1[3 * HID + k]);
        float cn = fv * c0b[k] + iv * gv;
        h2v[k] = ov * fast_tanh(cn);
    }

    // final linear + tanh
    float oacc = smem[OFF_BLIN];
#pragma unroll
    for (int k = 0; k < HID; ++k)
        oacc = fmaf(smem[OFF_WLIN + k], h2v[k], oacc);
    out[s] = fast_tanh(oacc);
}

extern "C" void kernel_launch(void* const* d_in, const int* in_sizes, int n_in,
                              void* d_out, int out_size, void* d_ws, size_t ws_size,
                              hipStream_t stream) {
    const float* x    = (const float*)d_in[0];
    const float* h0   = (const float*)d_in[1];
    const float* c0   = (const float*)d_in[2];
    const float* Wih0 = (const float*)d_in[3];
    const float* Whh0 = (const float*)d_in[4];
    const float* bih0 = (const float*)d_in[5];
    const float* bhh0 = (const float*)d_in[6];
    const float* Wih1 = (const float*)d_in[7];
    const float* Whh1 = (const float*)d_in[8];
    const float* bih1 = (const float*)d_in[9];
    const float* bhh1 = (const float*)d_in[10];
    const float* Wlin = (const float*)d_in[11];
    const float* blin = (const float*)d_in[12];
    float* out = (float*)d_out;

    const int Bn = in_sizes[0] / IN_DIM;     // 2^20
    const int blocks = Bn / 256;             // 256 samples per block (8 waves x 32)
    lstm2_wmma_kernel<<<blocks, 256, 0, stream>>>(
        x, h0, c0, Wih0, Whh0, bih0, bhh0,
        Wih1, Whh1, bih1, bhh1, Wlin, blin, out, Bn);
}